// DecoderWithAttention_19765439496562
// MI455X (gfx1250) — compile-verified
//
#include <hip/hip_runtime.h>
#include <math.h>

// Problem dimensions (from reference)
#define B_  64
#define P_  196
#define ED_ 512
#define A_  512
#define E_  512
#define D_  512
#define V_  5004
#define T_  64

#define NT 4   // 16-col C tiles per wave (A-fragment reuse factor)

typedef float v2f __attribute__((ext_vector_type(2)));
typedef float v8f __attribute__((ext_vector_type(8)));

// ---------------------------------------------------------------------------
// One product pass: acc[j] += A[M,K] @ Bm[K,N] for this wave's 16x(16*NT)
// strip. Per-lane B base pointers are loop-invariant (hoisted); only a
// uniform stride advance (4*N floats) happens per k-step. Prefetch is
// branchless so the loop stays one basic block and keeps the clause-grouped,
// software-pipelined schedule (loads overlap the 4 back-to-back WMMAs).
// ---------------------------------------------------------------------------
__device__ __forceinline__ void wmma_product(const float* __restrict__ Arow,
                                             const float* __restrict__ Bm,
                                             int K, int N,
                                             const int* colC, int half,
                                             v8f* acc)
{
    const size_t strideK4 = (size_t)4 * N;          // floats per k+=4
    const float* __restrict__ Bp[NT];
#pragma unroll
    for (int j = 0; j < NT; ++j)
        Bp[j] = Bm + (size_t)(half << 1) * N + colC[j];
    const float* __restrict__ Ap = Arow + (half << 1);

    for (int k = 0; k < K; k += 4) {
        v2f a;
        a.x = Ap[0];
        a.y = Ap[1];
        const int kp = (k + 4 < K) ? 4 : 0;          // branchless prefetch dist
        __builtin_prefetch(Bp[0] + (size_t)kp * N, 0, 2);
#pragma unroll
        for (int j = 0; j < NT; ++j) {
            v2f b;
            b.x = Bp[j][0];
            b.y = Bp[j][N];
            acc[j] = __builtin_amdgcn_wmma_f32_16x16x4_f32(
                false, a, false, b, (short)0, acc[j], false, false);
            Bp[j] += strideK4;
        }
        Ap += 4;
    }
}

// ---------------------------------------------------------------------------
// FP32 WMMA GEMM:
//   C[M,N] = A[M,K] @ Bm[K,N] (+ A2[M,K2] @ B2[K2,N]) (+ bias [+ bias2])
// with optional row mask (row active iff step < dec_len[row]; else 0).
// One wave32 computes a 16x(16*NT) strip of C via V_WMMA_F32_16X16X4_F32.
// Wave-uniform early exit keeps EXEC all-ones at every WMMA (ISA 7.12 req).
// ---------------------------------------------------------------------------
__global__ void gemm_wmma_f32_kernel(const float* __restrict__ A,
                                     const float* __restrict__ Bm, int K,
                                     const float* __restrict__ A2,
                                     const float* __restrict__ B2, int K2,
                                     const float* __restrict__ bias,
                                     const float* __restrict__ bias2,
                                     float* __restrict__ C,
                                     int M, int N, size_t ldc,
                                     const int* __restrict__ dec_len, int step)
{
    const int lane    = threadIdx.x & 31;
    const int wave    = threadIdx.x >> 5;
    const int tilesN  = (N + 15) >> 4;
    const int groupsN = (tilesN + NT - 1) / NT;
    const int tilesM  = (M + 15) >> 4;
    const int numTiles = tilesM * groupsN;
    const int tile = blockIdx.x * (blockDim.x >> 5) + wave;
    if (tile >= numTiles) return;              // uniform across the wave

    const int tM   = (tile / groupsN) << 4;
    const int tgN  = (tile % groupsN) * NT;    // base 16-col tile index
    const int half = lane >> 4;                // 0: K=k..k+1, 1: K=k+2..k+3
    const int l16  = lane & 15;

    int rowA = tM + l16; if (rowA >= M) rowA = M - 1;   // clamp, keep EXEC full

    int colB[NT], colC[NT];
#pragma unroll
    for (int j = 0; j < NT; ++j) {
        const int cb = ((tgN + j) << 4) + l16;
        colB[j] = cb;
        colC[j] = cb < N ? cb : N - 1;
    }

    v8f acc[NT] = {};

    wmma_product(A + (size_t)rowA * K, Bm, K, N, colC, half, acc);
    if (A2)   // optional second product, accumulated in registers
        wmma_product(A2 + (size_t)rowA * K2, B2, K2, N, colC, half, acc);

#pragma unroll
    for (int j = 0; j < NT; ++j) {
        if (colB[j] < N) {
            float bv = bias ? bias[colB[j]] : 0.0f;
            if (bias2) bv += bias2[colB[j]];
#pragma unroll
            for (int r = 0; r < 8; ++r) {
                const int row = tM + r + (half << 3);   // C layout: VGPR r -> M=r / r+8
                if (row < M) {
                    float v = acc[j][r] + bv;
                    if (dec_len && !(step < dec_len[row])) v = 0.0f;
                    C[(size_t)row * ldc + colB[j]] = v;
                }
            }
        }
    }
}

// ---------------------------------------------------------------------------
// Stable descending argsort of lens (B=64), caps gather, int->float outputs.
// ---------------------------------------------------------------------------
__global__ void sort_gather_kernel(const int* __restrict__ lens,     // (B,1)
                                   const int* __restrict__ caps,     // (B,T)
                                   int* __restrict__ order,
                                   int* __restrict__ dec_len,
                                   int* __restrict__ caps_sorted,
                                   float* __restrict__ out_caps,
                                   float* __restrict__ out_declen,
                                   float* __restrict__ out_order)
{
    const int i = threadIdx.x;
    if (i < B_) {
        const int li = lens[i];
        int pos = 0;
        for (int j = 0; j < B_; ++j) {
            const int lj = lens[j];
            if (lj > li || (lj == li && j < i)) pos++;   // stable descending
        }
        order[pos]   = i;
        dec_len[pos] = li - 1;
    }
    __syncthreads();
    if (i < B_) {
        const int src = order[i];
        for (int t = 0; t < T_; ++t) {
            const int v = caps[src * T_ + t];
            caps_sorted[i * T_ + t] = v;
            out_caps[i * T_ + t]    = (float)v;
        }
        out_declen[i] = (float)dec_len[i];
        out_order[i]  = (float)order[i];
    }
}

__global__ void gather_enc_kernel(const float* __restrict__ enc_in,
                                  const int* __restrict__ order,
                                  float* __restrict__ enc_out)
{
    const size_t n = (size_t)B_ * P_ * ED_;
    for (size_t i = (size_t)blockIdx.x * blockDim.x + threadIdx.x; i < n;
         i += (size_t)gridDim.x * blockDim.x) {
        const size_t b    = i / ((size_t)P_ * ED_);
        const size_t rest = i % ((size_t)P_ * ED_);
        enc_out[i] = enc_in[(size_t)order[b] * P_ * ED_ + rest];
    }
}

__global__ void mean_enc_kernel(const float* __restrict__ enc,
                                float* __restrict__ mean_enc)
{
    const int i = blockIdx.x * blockDim.x + threadIdx.x;
    if (i >= B_ * ED_) return;
    const int b = i / ED_, e = i % ED_;
    const float* base = enc + (size_t)b * P_ * ED_ + e;
    float s = 0.0f;
    for (int p = 0; p < P_; ++p) s += base[(size_t)p * ED_];
    mean_enc[i] = s * (1.0f / (float)P_);
}

__global__ void emb_gather_kernel(const float* __restrict__ emb,
                                  const int* __restrict__ caps_sorted,
                                  float* __restrict__ emb_seq)   // (B, T-1, E)
{
    const size_t n = (size_t)B_ * (T_ - 1) * E_;
    for (size_t i = (size_t)blockIdx.x * blockDim.x + threadIdx.x; i < n;
         i += (size_t)gridDim.x * blockDim.x) {
        const int    e  = (int)(i % E_);
        const size_t bt = i / E_;
        const int    t  = (int)(bt % (T_ - 1));
        const int    b  = (int)(bt / (T_ - 1));
        const int tok = caps_sorted[b * T_ + t];
        emb_seq[i] = emb[(size_t)tok * E_ + e];
    }
}

// ---------------------------------------------------------------------------
// Attention: att = relu(enc_att + hWda) . w_fa + b_fa; softmax over P;
// awe = sum_p alpha * enc. One block (256 thr) per batch row.
// ---------------------------------------------------------------------------
__global__ void attention_kernel(const float* __restrict__ enc_att,  // (B,P,A)
                                 const float* __restrict__ hWda,     // (B,A) incl b_da
                                 const float* __restrict__ w_fa,     // (A,)
                                 const float* __restrict__ b_fa_p,   // scalar
                                 const float* __restrict__ enc,      // (B,P,ED)
                                 const int* __restrict__ dec_len, int step,
                                 float* __restrict__ alpha_ws,       // (B,P)
                                 float* __restrict__ alphas_out,     // (B,T-1,P)
                                 float* __restrict__ awe)            // (B,ED)
{
    const int b   = blockIdx.x;
    const int tid = threadIdx.x;
    __shared__ float s_h[A_];
    __shared__ float s_red[256];
    __shared__ float s_alpha[256];

    for (int a = tid; a < A_; a += blockDim.x) s_h[a] = hWda[b * A_ + a];
    __syncthreads();

    float att = -3.0e38f;
    if (tid < P_) {
        float sum = b_fa_p[0];
        const float* row = enc_att + ((size_t)b * P_ + tid) * A_;
        for (int a = 0; a < A_; ++a) {
            float v = row[a] + s_h[a];
            v = v > 0.0f ? v : 0.0f;
            sum += v * w_fa[a];
        }
        att = sum;
    }
    s_red[tid] = att;
    __syncthreads();
    for (int s = 128; s > 0; s >>= 1) {
        if (tid < s) s_red[tid] = fmaxf(s_red[tid], s_red[tid + s]);
        __syncthreads();
    }
    const float m = s_red[0];
    __syncthreads();

    const float ex = (tid < P_) ? expf(att - m) : 0.0f;
    s_red[tid] = ex;
    __syncthreads();
    for (int s = 128; s > 0; s >>= 1) {
        if (tid < s) s_red[tid] += s_red[tid + s];
        __syncthreads();
    }
    const float Z = s_red[0];
    const float alpha = ex / Z;
    const bool active = step < dec_len[b];
    if (tid < P_) {
        alpha_ws[b * P_ + tid] = alpha;
        alphas_out[((size_t)b * (T_ - 1) + step) * P_ + tid] = active ? alpha : 0.0f;
    }
    s_alpha[tid] = (tid < P_) ? alpha : 0.0f;
    __syncthreads();

    for (int e = tid; e < ED_; e += blockDim.x) {
        const float* base = enc + (size_t)b * P_ * ED_ + e;
        float s = 0.0f;
        for (int p = 0; p < P_; ++p) s += s_alpha[p] * base[(size_t)p * ED_];
        awe[b * ED_ + e] = s;
    }
}

__global__ void build_xin_kernel(const float* __restrict__ emb_seq, int step,
                                 const float* __restrict__ gateLin,  // (B,ED) pre-sigmoid
                                 const float* __restrict__ awe,      // (B,ED)
                                 float* __restrict__ xin)            // (B, E+ED)
{
    const int i = blockIdx.x * blockDim.x + threadIdx.x;
    if (i >= B_ * ED_) return;
    const int b = i / ED_, e = i % ED_;
    xin[(size_t)b * (E_ + ED_) + e] = emb_seq[((size_t)b * (T_ - 1) + step) * E_ + e];
    const float g = 1.0f / (1.0f + expf(-gateLin[i]));
    xin[(size_t)b * (E_ + ED_) + E_ + e] = g * awe[i];
}

__global__ void lstm_cell_kernel(const float* __restrict__ g,   // (B,4D) fused gates
                                 float* __restrict__ h,
                                 float* __restrict__ c,
                                 float* __restrict__ h_new,
                                 const int* __restrict__ dec_len, int step)
{
    const int i = blockIdx.x * blockDim.x + threadIdx.x;
    if (i >= B_ * D_) return;
    const int b = i / D_, j = i % D_;
    const float* gb = g + (size_t)b * 4 * D_;
    const float ig = gb[j];
    const float fg = gb[D_ + j];
    const float gg = gb[2 * D_ + j];
    const float og = gb[3 * D_ + j];
    const float si = 1.0f / (1.0f + expf(-ig));
    const float sf = 1.0f / (1.0f + expf(-fg));
    const float so = 1.0f / (1.0f + expf(-og));
    const float cn = sf * c[i] + si * tanhf(gg);
    const float hn = so * tanhf(cn);
    h_new[i] = hn;
    if (step < dec_len[b]) { h[i] = hn; c[i] = cn; }
}

// ---------------------------------------------------------------------------
static inline void* carve(char*& p, size_t bytes) {
    void* r = (void*)p;
    p += (bytes + 255) & ~(size_t)255;
    return r;
}

extern "C" void kernel_launch(void* const* d_in, const int* in_sizes, int n_in,
                              void* d_out, int out_size, void* d_ws, size_t ws_size,
                              hipStream_t stream)
{
    (void)in_sizes; (void)n_in; (void)out_size; (void)ws_size;

    const float* encoder_out      = (const float*)d_in[0];
    const int*   encoded_captions = (const int*)  d_in[1];
    const int*   caption_lengths  = (const int*)  d_in[2];
    const float* emb    = (const float*)d_in[3];
    const float* W_ea   = (const float*)d_in[4];
    const float* b_ea   = (const float*)d_in[5];
    const float* W_da   = (const float*)d_in[6];
    const float* b_da   = (const float*)d_in[7];
    const float* w_fa   = (const float*)d_in[8];
    const float* b_fa   = (const float*)d_in[9];
    const float* W_ih   = (const float*)d_in[10];
    const float* W_hh   = (const float*)d_in[11];
    const float* b_ih   = (const float*)d_in[12];
    const float* b_hh   = (const float*)d_in[13];
    const float* W_h0   = (const float*)d_in[14];
    const float* b_h0   = (const float*)d_in[15];
    const float* W_c0   = (const float*)d_in[16];
    const float* b_c0   = (const float*)d_in[17];
    const float* W_beta = (const float*)d_in[18];
    const float* b_beta = (const float*)d_in[19];
    const float* W_fc   = (const float*)d_in[20];
    const float* b_fc   = (const float*)d_in[21];

    // Output layout: predictions | alphas | caps | dec_len | order (all float)
    float* out        = (float*)d_out;
    float* out_pred   = out;
    float* out_alphas = out_pred   + (size_t)B_ * (T_ - 1) * V_;
    float* out_caps   = out_alphas + (size_t)B_ * (T_ - 1) * P_;
    float* out_declen = out_caps   + (size_t)B_ * T_;
    float* out_order  = out_declen + B_;

    // Workspace
    char* wsp = (char*)d_ws;
    int*   order_i     = (int*)  carve(wsp, sizeof(int) * B_);
    int*   declen_i    = (int*)  carve(wsp, sizeof(int) * B_);
    int*   caps_i      = (int*)  carve(wsp, sizeof(int) * B_ * T_);
    float* enc_sorted  = (float*)carve(wsp, sizeof(float) * (size_t)B_ * P_ * ED_);
    float* enc_att     = (float*)carve(wsp, sizeof(float) * (size_t)B_ * P_ * A_);
    float* mean_enc    = (float*)carve(wsp, sizeof(float) * B_ * ED_);
    float* emb_seq     = (float*)carve(wsp, sizeof(float) * (size_t)B_ * (T_ - 1) * E_);
    float* h           = (float*)carve(wsp, sizeof(float) * B_ * D_);
    float* c           = (float*)carve(wsp, sizeof(float) * B_ * D_);
    float* h_new       = (float*)carve(wsp, sizeof(float) * B_ * D_);
    float* hWda        = (float*)carve(wsp, sizeof(float) * B_ * A_);
    float* gateLin     = (float*)carve(wsp, sizeof(float) * B_ * ED_);
    float* awe         = (float*)carve(wsp, sizeof(float) * B_ * ED_);
    float* alpha_ws    = (float*)carve(wsp, sizeof(float) * B_ * P_);
    float* xin         = (float*)carve(wsp, sizeof(float) * B_ * (E_ + ED_));
    float* g_buf       = (float*)carve(wsp, sizeof(float) * B_ * 4 * D_);

    auto gemm = [&](const float* A, const float* Bm, int K,
                    const float* A2, const float* B2, int K2,
                    const float* bias, const float* bias2, float* C,
                    int M, int N, size_t ldc, const int* dl, int step) {
        const int tilesN  = (N + 15) / 16;
        const int groupsN = (tilesN + NT - 1) / NT;
        const int tiles   = ((M + 15) / 16) * groupsN;
        const int blocks  = (tiles + 7) / 8;          // 8 waves (256 thr) per block
        gemm_wmma_f32_kernel<<<blocks, 256, 0, stream>>>(A, Bm, K, A2, B2, K2,
                                                         bias, bias2, C, M, N,
                                                         ldc, dl, step);
    };

    // ---- setup ----
    sort_gather_kernel<<<1, 64, 0, stream>>>(caption_lengths, encoded_captions,
                                             order_i, declen_i, caps_i,
                                             out_caps, out_declen, out_order);
    {
        const size_t n = (size_t)B_ * P_ * ED_;
        gather_enc_kernel<<<(int)((n + 255) / 256), 256, 0, stream>>>(encoder_out, order_i, enc_sorted);
    }
    mean_enc_kernel<<<(B_ * ED_ + 255) / 256, 256, 0, stream>>>(enc_sorted, mean_enc);
    {
        const size_t n = (size_t)B_ * (T_ - 1) * E_;
        emb_gather_kernel<<<(int)((n + 255) / 256), 256, 0, stream>>>(emb, caps_i, emb_seq);
    }
    gemm(mean_enc,   W_h0, ED_, nullptr, nullptr, 0, b_h0, nullptr, h,       B_,      D_, D_, nullptr, 0);
    gemm(mean_enc,   W_c0, ED_, nullptr, nullptr, 0, b_c0, nullptr, c,       B_,      D_, D_, nullptr, 0);
    gemm(enc_sorted, W_ea, ED_, nullptr, nullptr, 0, b_ea, nullptr, enc_att, B_ * P_, A_, A_, nullptr, 0);

    // ---- recurrent steps ----
    for (int t = 0; t < T_ - 1; ++t) {
        gemm(h, W_da, D_, nullptr, nullptr, 0, b_da, nullptr, hWda, B_, A_, A_, nullptr, 0);
        attention_kernel<<<B_, 256, 0, stream>>>(enc_att, hWda, w_fa, b_fa,
                                                 enc_sorted, declen_i, t,
                                                 alpha_ws, out_alphas, awe);
        gemm(h, W_beta, D_, nullptr, nullptr, 0, b_beta, nullptr, gateLin, B_, ED_, ED_, nullptr, 0);
        build_xin_kernel<<<(B_ * ED_ + 255) / 256, 256, 0, stream>>>(emb_seq, t, gateLin, awe, xin);
        // fused: g = xin@W_ih + h@W_hh + b_ih + b_hh (accumulated in registers)
        gemm(xin, W_ih, E_ + ED_, h, W_hh, D_, b_ih, b_hh, g_buf,
             B_, 4 * D_, 4 * D_, nullptr, 0);
        lstm_cell_kernel<<<(B_ * D_ + 255) / 256, 256, 0, stream>>>(g_buf, h, c,
                                                                    h_new, declen_i, t);
        // predictions[:, t, :] with inactive rows zeroed; ldc = (T-1)*V
        gemm(h_new, W_fc, D_, nullptr, nullptr, 0, b_fc, nullptr,
             out_pred + (size_t)t * V_, B_, V_, (size_t)(T_ - 1) * V_, declen_i, t);
    }
}